// SAM_7215545057811
// MI455X (gfx1250) — compile-verified
//
#include <hip/hip_runtime.h>
#include <stdint.h>

// ---------------- problem constants ----------------
#define BATCH    8
#define HW_      (1536 * 1536)           // 2359296 = 2304 * 1024
#define O_PAD    64                      // padded object slots (real: 50)
#define NOBJ_MAX 50
#define TILE     1024                    // pixels per block-iteration
#define BLOCK    256                     // 8 wave32 per workgroup
#define NSUB     8                       // one LDS sub-histogram per wave
#define G_HIST   192                     // blocks per batch, pass 1 (2304/192 = 12 iters)
#define G_LOSS   192                     // blocks per batch, pass 2
#define N_PART   (G_LOSS * BATCH)        // 1536 partial sums
#define N_TOTAL  18874368.0f             // B*C*H*W

// workspace layout (float offsets)
#define WS_CNT   0                        // [BATCH][O_PAD] counts
#define WS_SUM   (BATCH * O_PAD)          // [BATCH][O_PAD] pred sums
#define WS_GMAX  (2 * BATCH * O_PAD)      // 1 int
#define WS_PART  1040                     // N_PART floats

// ---------------- feature detection ----------------
#if defined(__has_builtin)
#  if __has_builtin(__builtin_amdgcn_global_load_async_to_lds_b128)
#    define HAVE_ASYNC_BUILTIN 1
#  endif
#  if __has_builtin(__builtin_amdgcn_s_wait_asynccnt)
#    define HAVE_WAIT_BUILTIN 1
#  endif
#  if __has_builtin(__builtin_amdgcn_wmma_f32_16x16x4_f32)
#    define HAVE_WMMA_F32 1
#  endif
#endif
#ifndef HAVE_ASYNC_BUILTIN
#define HAVE_ASYNC_BUILTIN 0
#endif
#ifndef HAVE_WAIT_BUILTIN
#define HAVE_WAIT_BUILTIN 0
#endif
#ifndef HAVE_WMMA_F32
#define HAVE_WMMA_F32 0
#endif

typedef __attribute__((ext_vector_type(4))) int   v4i;
typedef __attribute__((ext_vector_type(2))) float v2f;
typedef __attribute__((ext_vector_type(8))) float v8f;

// 16B async copy global -> LDS (per lane). Tracked by ASYNCcnt (CDNA5).
__device__ __forceinline__ void async_b128(const void* gsrc, void* ldst) {
#if HAVE_ASYNC_BUILTIN
  __builtin_amdgcn_global_load_async_to_lds_b128((v4i*)(void*)gsrc,
                                                 (v4i*)ldst, 0, 0);
#else
  unsigned loff = (unsigned)(uintptr_t)ldst;  // LDS-relative byte offset
  unsigned long long ga = (unsigned long long)(uintptr_t)gsrc;
  asm volatile("global_load_async_to_lds_b128 %0, %1, off"
               :: "v"(loff), "v"(ga) : "memory");
#endif
}

template <int N>
__device__ __forceinline__ void wait_async() {
#if HAVE_WAIT_BUILTIN
  __builtin_amdgcn_s_wait_asynccnt(N);
#else
  asm volatile("s_wait_asynccnt %0" :: "i"(N) : "memory");
#endif
  asm volatile("" ::: "memory");  // compiler barrier: no LDS-read hoisting
}

// ---------------- kernel 0: zero persistent stats ----------------
__global__ void sam_init(float* __restrict__ cnt_sum, int* __restrict__ gmax,
                         float* __restrict__ out) {
  int i = blockIdx.x * blockDim.x + threadIdx.x;
  if (i < 2 * BATCH * O_PAD) cnt_sum[i] = 0.0f;
  if (i == 0) { gmax[0] = 0; out[0] = 0.0f; }
}

// ---------------- kernel 1: per-(batch,object) counts & sums ----------------
__global__ __launch_bounds__(BLOCK)
void sam_hist(const float* __restrict__ pred, const int* __restrict__ gt,
              float* __restrict__ cnt, float* __restrict__ sums,
              int* __restrict__ gmax) {
  __shared__ float s_hist[NSUB][2 * O_PAD];  // [sub][0..63]=cnt, [64..127]=sum
  __shared__ float s_pred[2][TILE];
  __shared__ int   s_gt[2][TILE];
  __shared__ int   s_max;

  const int tid = threadIdx.x;
  const int wv  = tid >> 5;       // wave id: private sub-histogram
  const int b   = blockIdx.y;
  const float* predB = pred + (size_t)b * HW_;
  const int*   gtB   = gt   + (size_t)b * HW_;

  for (int i = tid; i < NSUB * 2 * O_PAD; i += BLOCK) (&s_hist[0][0])[i] = 0.0f;
  if (tid == 0) s_max = 0;
  __syncthreads();

  const int T  = HW_ / TILE;      // 2304 tiles per batch
  const int t4 = tid * 4;         // this thread's 4-float slot in the tile
  int localmax = 0;

  // prologue: issue tile 0 into buffer 0 (each wave loads only its own slice)
  int tile = blockIdx.x;
  async_b128(predB + (size_t)tile * TILE + t4, &s_pred[0][t4]);
  async_b128(gtB   + (size_t)tile * TILE + t4, &s_gt[0][t4]);

  int buf = 0;
  for (; tile < T; tile += G_HIST) {
    const int next = tile + G_HIST;
    if (next < T) {
      async_b128(predB + (size_t)next * TILE + t4, &s_pred[buf ^ 1][t4]);
      async_b128(gtB   + (size_t)next * TILE + t4, &s_gt[buf ^ 1][t4]);
      wait_async<2>();            // current tile's 2 loads done; next 2 in flight
    } else {
      wait_async<0>();
    }
    const float4 p = *(const float4*)&s_pred[buf][t4];
    const int4   g = *(const int4*)&s_gt[buf][t4];
    const int g0 = g.x & 63, g1 = g.y & 63, g2 = g.z & 63, g3 = g.w & 63;
    float* h = s_hist[wv];
    atomicAdd(&h[g0], 1.0f); atomicAdd(&h[O_PAD + g0], p.x);
    atomicAdd(&h[g1], 1.0f); atomicAdd(&h[O_PAD + g1], p.y);
    atomicAdd(&h[g2], 1.0f); atomicAdd(&h[O_PAD + g2], p.z);
    atomicAdd(&h[g3], 1.0f); atomicAdd(&h[O_PAD + g3], p.w);
    localmax = max(max(localmax, max(g.x, g.y)), max(g.z, g.w));
    buf ^= 1;
  }

  atomicMax(&s_max, localmax);
  __syncthreads();

  // merge sub-histograms, accumulate to global stats
  for (int i = tid; i < 2 * O_PAD; i += BLOCK) {
    float v = 0.0f;
#pragma unroll
    for (int s = 0; s < NSUB; ++s) v += s_hist[s][i];
    if (i < O_PAD) atomicAdd(&cnt[b * O_PAD + i], v);
    else           atomicAdd(&sums[b * O_PAD + (i - O_PAD)], v);
  }
  if (tid == 0) atomicMax(gmax, s_max);
}

// ---------------- kernel 2: per-pixel squared loss -> per-block partials ----
__global__ __launch_bounds__(BLOCK)
void sam_loss(const float* __restrict__ pred, const int* __restrict__ gt,
              const float* __restrict__ cnt, const float* __restrict__ sums,
              const int* __restrict__ gmax, float* __restrict__ partials) {
  __shared__ float s_cnt[O_PAD];
  __shared__ float s_sum[O_PAD];
  __shared__ float s_pred[2][TILE];
  __shared__ int   s_gt[2][TILE];
  __shared__ float s_red[NSUB];

  const int tid = threadIdx.x;
  const int b   = blockIdx.y;
  if (tid < O_PAD)          s_cnt[tid] = cnt[b * O_PAD + tid];
  else if (tid < 2 * O_PAD) s_sum[tid - O_PAD] = sums[b * O_PAD + tid - O_PAD];
  const int nobj = min(gmax[0] + 1, NOBJ_MAX);
  __syncthreads();

  const float* predB = pred + (size_t)b * HW_;
  const int*   gtB   = gt   + (size_t)b * HW_;
  const int T  = HW_ / TILE;
  const int t4 = tid * 4;
  float acc = 0.0f;

  int tile = blockIdx.x;
  async_b128(predB + (size_t)tile * TILE + t4, &s_pred[0][t4]);
  async_b128(gtB   + (size_t)tile * TILE + t4, &s_gt[0][t4]);

  int buf = 0;
  for (; tile < T; tile += G_LOSS) {
    const int next = tile + G_LOSS;
    if (next < T) {
      async_b128(predB + (size_t)next * TILE + t4, &s_pred[buf ^ 1][t4]);
      async_b128(gtB   + (size_t)next * TILE + t4, &s_gt[buf ^ 1][t4]);
      wait_async<2>();
    } else {
      wait_async<0>();
    }
    const float4 p = *(const float4*)&s_pred[buf][t4];
    const int4   g = *(const int4*)&s_gt[buf][t4];
#pragma unroll
    for (int j = 0; j < 4; ++j) {
      const int   gi = (j == 0) ? g.x : (j == 1) ? g.y : (j == 2) ? g.z : g.w;
      const float pi = (j == 0) ? p.x : (j == 1) ? p.y : (j == 2) ? p.z : p.w;
      const int   o  = gi & 63;
      const float np = s_cnt[o];
      const float S  = s_sum[o];
      const float d  = (pi * np - S) / (np + 1.0f);
      acc += (gi >= 1 && gi < nobj) ? d * d : 0.0f;
    }
    buf ^= 1;
  }

  // deterministic block reduction
#pragma unroll
  for (int off = 16; off > 0; off >>= 1) acc += __shfl_down(acc, off);
  if ((tid & 31) == 0) s_red[tid >> 5] = acc;
  __syncthreads();
  if (tid == 0) {
    float t = 0.0f;
#pragma unroll
    for (int s = 0; s < NSUB; ++s) t += s_red[s];
    partials[b * G_LOSS + blockIdx.x] = t;
  }
}

// ---------------- kernel 3: fixed-order final reduction (WMMA row-sum) -----
__global__ __launch_bounds__(BLOCK)
void sam_final(const float* __restrict__ partials, float* __restrict__ out) {
  __shared__ float s_v[BLOCK];
  __shared__ float s_w[64];
  const int tid = threadIdx.x;
  float acc = 0.0f;
  for (int i = tid; i < N_PART; i += BLOCK) acc += partials[i];
  s_v[tid] = acc;
  __syncthreads();
  if (tid < 64) s_w[tid] = s_v[tid] + s_v[tid + 64] + s_v[tid + 128] + s_v[tid + 192];
  __syncthreads();
  if (tid < 32) {  // wave 0, EXEC all ones for WMMA
    float tot;
#if HAVE_WMMA_F32
    // A = 16x4 f32 holding the 64 values (A[m][k] = s_w[4m+k]); B = ones(4x16)
    // => D[m][n] = rowsum[m]. Lane 0 holds rowsums 0..7, lane 16 holds 8..15.
    v2f a;
    if (tid < 16) { a.x = s_w[tid * 4 + 0];        a.y = s_w[tid * 4 + 1]; }
    else          { a.x = s_w[(tid - 16) * 4 + 2]; a.y = s_w[(tid - 16) * 4 + 3]; }
    v2f bb; bb.x = 1.0f; bb.y = 1.0f;
    v8f c = {};
    v8f dm = __builtin_amdgcn_wmma_f32_16x16x4_f32(false, a, false, bb,
                                                   (short)0, c, false, false);
    float s = dm[0] + dm[1] + dm[2] + dm[3] + dm[4] + dm[5] + dm[6] + dm[7];
    tot = s + __shfl_xor(s, 16);
#else
    tot = 0.0f;
    if (tid == 0) { for (int i = 0; i < 64; ++i) tot += s_w[i]; }
#endif
    if (tid == 0) out[0] = tot * (1.0f / N_TOTAL);
  }
}

// ---------------- host entry ----------------
extern "C" void kernel_launch(void* const* d_in, const int* in_sizes, int n_in,
                              void* d_out, int out_size, void* d_ws, size_t ws_size,
                              hipStream_t stream) {
  const float* pred = (const float*)d_in[0];
  const int*   gt   = (const int*)d_in[1];
  float* ws       = (float*)d_ws;
  float* cnt      = ws + WS_CNT;
  float* sums     = ws + WS_SUM;
  int*   gmax     = (int*)(ws + WS_GMAX);
  float* partials = ws + WS_PART;
  float* out      = (float*)d_out;

  sam_init<<<4, BLOCK, 0, stream>>>(cnt, gmax, out);
  sam_hist<<<dim3(G_HIST, BATCH), BLOCK, 0, stream>>>(pred, gt, cnt, sums, gmax);
  sam_loss<<<dim3(G_LOSS, BATCH), BLOCK, 0, stream>>>(pred, gt, cnt, sums, gmax, partials);
  sam_final<<<1, BLOCK, 0, stream>>>(partials, out);
}